// GraphODEFunc_7035156431297
// MI455X (gfx1250) — compile-verified
//
#include <hip/hip_runtime.h>
#include <hip/hip_bf16.h>
#include <math.h>

typedef float v2f __attribute__((ext_vector_type(2)));
typedef float v8f __attribute__((ext_vector_type(8)));

#define BB 4
#define DD 64

// ---------------------------------------------------------------------------
// Degree / normalization
// ---------------------------------------------------------------------------
__global__ void deg_init(unsigned int* deg, int N) {
    int i = blockIdx.x * 256 + threadIdx.x;
    if (i < N) deg[i] = 1u;  // self-loop contributes 1
}

__global__ void deg_count(const int* __restrict__ coli, unsigned int* deg, int E) {
    int e = blockIdx.x * 256 + threadIdx.x;
    if (e < E) atomicAdd(&deg[coli[e]], 1u);
}

__global__ void deg_rsqrt(unsigned int* __restrict__ deg, float* __restrict__ dis, int N) {
    int i = blockIdx.x * 256 + threadIdx.x;
    if (i < N) {
        float f = (float)deg[i];
        dis[i] = rsqrtf(f);  // deg >= 1 always (self loop)
    }
}

// ---------------------------------------------------------------------------
// Dense 64x64 transform via V_WMMA_F32_16X16X4_F32.
// One block = 16 rows; wave w (of 4) computes output columns [16w, 16w+16).
// A layout (16x4 f32): lanes 0-15 -> M=lane, K={0,1}; lanes 16-31 -> M=lane-16, K={2,3}
// B layout (4x16 f32): lanes 0-15 -> N=lane, K={0,1}; lanes 16-31 -> N=lane-16, K={2,3}
// C/D layout: vgpr r, lanes 0-15 -> (M=r, N=lane); lanes 16-31 -> (M=r+8, N=lane-16)
// ---------------------------------------------------------------------------
__global__ __launch_bounds__(128) void gemm64(const float* __restrict__ X,
                                              const float* __restrict__ W,
                                              float* __restrict__ Y, int rows) {
    const int rowbase = blockIdx.x * 16;
    const int wave    = threadIdx.x >> 5;
    const int lane    = threadIdx.x & 31;
    const int colbase = wave * 16;
    const int mn      = lane & 15;          // M for A, N for B/C
    const int kb      = (lane >> 4) * 2;    // 0 or 2

    int ri = rowbase + mn;
    if (ri >= rows) ri = rows - 1;          // clamp (rows is a multiple of 16 normally)
    const float* xrow = X + (size_t)ri * DD;

    v8f c = {0.f, 0.f, 0.f, 0.f, 0.f, 0.f, 0.f, 0.f};
#pragma unroll
    for (int k0 = 0; k0 < DD; k0 += 4) {
        v2f a, b;
        a.x = xrow[k0 + kb];
        a.y = xrow[k0 + kb + 1];
        b.x = W[(size_t)(k0 + kb) * DD + colbase + mn];
        b.y = W[(size_t)(k0 + kb + 1) * DD + colbase + mn];
        c = __builtin_amdgcn_wmma_f32_16x16x4_f32(false, a, false, b,
                                                  (short)0, c, false, false);
    }

    const int mofs = (lane >> 4) * 8;
#pragma unroll
    for (int r = 0; r < 8; ++r) {
        int ro = rowbase + mofs + r;
        if (ro < rows) Y[(size_t)ro * DD + colbase + mn] = c[r];
    }
}

// ---------------------------------------------------------------------------
// agg = dis[n]^2 * xw   (self-loop term; also initializes the accumulator)
// One thread per float4 (16 float4s per row of 64).
// ---------------------------------------------------------------------------
__global__ __launch_bounds__(256) void self_init(const float* __restrict__ dis,
                                                 const float* __restrict__ xw,
                                                 float* __restrict__ agg,
                                                 int N, long total4) {
    long u = (long)blockIdx.x * 256 + threadIdx.x;
    if (u >= total4) return;
    long bn = u >> 4;              // flat (b*N + n)
    int  n  = (int)(bn % N);
    float s  = dis[n];
    float s2 = s * s;
    float4 v = ((const float4*)xw)[u];
    float4 o;
    o.x = s2 * v.x; o.y = s2 * v.y; o.z = s2 * v.z; o.w = s2 * v.w;
    ((float4*)agg)[u] = o;
}

// ---------------------------------------------------------------------------
// Hardware fp32 atomic add, guaranteed: CDNA5 GLOBAL_ATOMIC_ADD_F32, no-return
// form (STOREcnt), device scope so the RMW resolves at L2 and is coherent
// across WGPs. Avoids any chance of a CAS-loop expansion in the hot scatter.
// ---------------------------------------------------------------------------
__device__ __forceinline__ void fadd_atomic(float* p, float v) {
    asm volatile("global_atomic_add_f32 %0, %1, off scope:SCOPE_DEV"
                 :
                 : "v"(p), "v"(v)
                 : "memory");
}

// ---------------------------------------------------------------------------
// Edge scatter: agg[b, col[e], :] += dis[row]*dis[col] * xw[b, row[e], :]
// One wave per edge; lane owns features {2*lane, 2*lane+1}; loop over B=4.
// Accumulator (51.2 MB) is L2-resident -> fp32 atomics run near L2 rate.
// ---------------------------------------------------------------------------
__global__ __launch_bounds__(256) void scatter(const int* __restrict__ rowi,
                                               const int* __restrict__ coli,
                                               const float* __restrict__ dis,
                                               const float* __restrict__ xw,
                                               float* __restrict__ agg,
                                               int E, int N) {
    int e = blockIdx.x * 8 + (threadIdx.x >> 5);
    if (e >= E) return;
    int lane = threadIdx.x & 31;
    int r = rowi[e];
    int c = coli[e];
    float nrm = dis[r] * dis[c];
    const size_t ND = (size_t)N * DD;
#pragma unroll
    for (int b = 0; b < BB; ++b) {
        float2 v = *(const float2*)(xw + (size_t)b * ND + (size_t)r * DD + lane * 2);
        float* dst = agg + (size_t)b * ND + (size_t)c * DD + lane * 2;
        fadd_atomic(dst + 0, nrm * v.x);
        fadd_atomic(dst + 1, nrm * v.y);
    }
}

// ---------------------------------------------------------------------------
// out = tanh(agg + bias[d])
// ---------------------------------------------------------------------------
__global__ __launch_bounds__(256) void bias_tanh(const float* __restrict__ agg,
                                                 const float* __restrict__ bias,
                                                 float* __restrict__ out, long total) {
    long i = (long)blockIdx.x * 256 + threadIdx.x;
    if (i >= total) return;
    int d = (int)(i & (DD - 1));
    out[i] = tanhf(agg[i] + bias[d]);
}

// ---------------------------------------------------------------------------
// Launch
// d_in: 0=t, 1=h(B,N,D), 2=edge_index(2,E), 3=W1, 4=b1, 5=W2, 6=b2
// ---------------------------------------------------------------------------
extern "C" void kernel_launch(void* const* d_in, const int* in_sizes, int n_in,
                              void* d_out, int out_size, void* d_ws, size_t ws_size,
                              hipStream_t stream) {
    const float* h  = (const float*)d_in[1];
    const int*   ei = (const int*)d_in[2];
    const float* W1 = (const float*)d_in[3];
    const float* b1 = (const float*)d_in[4];
    const float* W2 = (const float*)d_in[5];
    const float* b2 = (const float*)d_in[6];
    float* out = (float*)d_out;

    const int  E      = in_sizes[2] / 2;
    const int  N      = in_sizes[1] / (BB * DD);
    const int  rows   = BB * N;
    const long total  = (long)rows * DD;
    const long total4 = total / 4;

    const int* rowi = ei;
    const int* coli = ei + E;

    // workspace: [deg/dis: N floats (aligned)] [bufA: total floats] [bufB: total floats]
    unsigned int* deg = (unsigned int*)d_ws;
    float*        dis = (float*)d_ws;
    size_t nAl = ((size_t)N + 1023) & ~(size_t)1023;
    float* bufA = (float*)d_ws + nAl;
    float* bufB = bufA + total;

    const int tpb = 256;
    int gN  = (N + tpb - 1) / tpb;
    int gE  = (E + tpb - 1) / tpb;
    int gT  = (int)((total + tpb - 1) / tpb);
    int gT4 = (int)((total4 + tpb - 1) / tpb);
    int gG  = (rows + 15) / 16;
    int gS  = (E + 7) / 8;

    // normalization
    deg_init<<<gN, tpb, 0, stream>>>(deg, N);
    deg_count<<<gE, tpb, 0, stream>>>(coli, deg, E);
    deg_rsqrt<<<gN, tpb, 0, stream>>>(deg, dis, N);

    // ---- layer 1 ----
    gemm64<<<gG, 128, 0, stream>>>(h, W1, bufA, rows);                 // xw1 -> A
    self_init<<<gT4, tpb, 0, stream>>>(dis, bufA, bufB, N, total4);    // agg1 -> B
    scatter<<<gS, tpb, 0, stream>>>(rowi, coli, dis, bufA, bufB, E, N);
    bias_tanh<<<gT, tpb, 0, stream>>>(bufB, b1, bufA, total);          // h1 -> A

    // ---- layer 2 ----
    gemm64<<<gG, 128, 0, stream>>>(bufA, W2, bufB, rows);              // xw2 -> B
    self_init<<<gT4, tpb, 0, stream>>>(dis, bufB, bufA, N, total4);    // agg2 -> A
    scatter<<<gS, tpb, 0, stream>>>(rowi, coli, dis, bufB, bufA, E, N);
    bias_tanh<<<gT, tpb, 0, stream>>>(bufA, b2, out, total);           // out
}